// Network_loss_38620345926210
// MI455X (gfx1250) — compile-verified
//
#include <hip/hip_runtime.h>
#include <math.h>

typedef __attribute__((ext_vector_type(2))) float v2f;
typedef __attribute__((ext_vector_type(4))) float v4f;
typedef __attribute__((ext_vector_type(8))) float v8f;

#define NU 8192
#define NI 8192
#define LOG1000F 6.907755278982137f
#define LN2F     0.6931471805599453f
// -100 / ln(2): BCE clamp bound expressed in log2 domain
#define NEG100_OVER_LN2 (-144.26950408889634f)

// BCE launch geometry: 2048 blocks x 256 threads = 2^19 threads,
// N4 = 8192*8192/4 = 2^24 float4 elements -> exactly 32 iterations, no tail.
#define BCE_BLOCKS   2048
#define BCE_THREADS  256
#define BCE_STRIDE   (BCE_BLOCKS * BCE_THREADS)   // 524288
#define BCE_ITERS    32

// raw hardware v_log_f32 (log2). Inputs are guaranteed in [1e-7, 1):
// no denormals / zeros / infs, so the OCML guard path is unnecessary.
__device__ __forceinline__ float fast_log2(float x) {
#if __has_builtin(__builtin_amdgcn_logf)
    return __builtin_amdgcn_logf(x);
#else
    return __logf(x) * 1.4426950408889634f;   // fallback: ln -> log2
#endif
}

// ws layout (floats):
// [0]      : 16x BCE term sum, log2 domain (WMMA ones-B replication)
// [1..16]  : column sums of log_p_user
// [17..32] : column sums of log_q_item
// [33]     : sum of log(alpha_item) over all elements
// [34]     : beta-prime sum (users + items)

__global__ void k_init(float* __restrict__ ws) {
    if (threadIdx.x < 40) ws[threadIdx.x] = 0.0f;
}

// ---- dominant kernel: BCE sum; raw v_log_f32 + WMMA f32 16x16x4 reduction ----
__global__ void __launch_bounds__(BCE_THREADS)
k_bce(const float* __restrict__ pred, const float* __restrict__ tgt,
      float* __restrict__ acc) {
    const unsigned tid = blockIdx.x * BCE_THREADS + threadIdx.x;

    v8f c = {0.f, 0.f, 0.f, 0.f, 0.f, 0.f, 0.f, 0.f};
    v2f bones; bones[0] = 1.0f; bones[1] = 1.0f;   // B = ones(4x16)

#pragma unroll 4
    for (int it = 0; it < BCE_ITERS; ++it) {
        const size_t i = ((size_t)it * BCE_STRIDE + tid) * 4u;
        v4f p4 = __builtin_nontemporal_load((const v4f*)(pred + i));
        v4f t4 = __builtin_nontemporal_load((const v4f*)(tgt  + i));
        float tm[4];
#pragma unroll
        for (int j = 0; j < 4; ++j) {
            float p    = fminf(p4[j], 0.9999999f);
            // log2-domain: max(ln x,-100) == ln2 * max(log2 x, -100/ln2)
            // 1-p exact for p>=0.5 (Sterbenz); fine below that for p>=1e-3
            float lp2  = fmaxf(fast_log2(p),        NEG100_OVER_LN2);
            float l1p2 = fmaxf(fast_log2(1.0f - p), NEG100_OVER_LN2);
            tm[j] = l1p2 + t4[j] * (lp2 - l1p2);    // == t*lp + (1-t)*l1p for t in {0,1}
        }
        v2f a0; a0[0] = tm[0]; a0[1] = tm[1];
        v2f a1; a1[0] = tm[2]; a1[1] = tm[3];
        // D = A x ones + C : XDL pipe does the adds, co-executes with TRANS logs
        c = __builtin_amdgcn_wmma_f32_16x16x4_f32(false, a0, false, bones, (short)0, c, false, false);
        c = __builtin_amdgcn_wmma_f32_16x16x4_f32(false, a1, false, bones, (short)0, c, false, false);
    }

    float s = 0.f;
#pragma unroll
    for (int j = 0; j < 8; ++j) s += c[j];
#pragma unroll
    for (int off = 16; off >= 1; off >>= 1) s += __shfl_xor(s, off, 32);

    __shared__ float red[8];
    if ((threadIdx.x & 31u) == 0u) red[threadIdx.x >> 5] = s;
    __syncthreads();
    if (threadIdx.x == 0) {
        float b = 0.f;
#pragma unroll
        for (int j = 0; j < 8; ++j) b += red[j];
        atomicAdd(acc, b);   // 16x the true log2-domain sum (ones-B replication)
    }
}

// ---- column sums of an [8192,16] matrix ----
__global__ void k_colsum16(const float* __restrict__ src, float* __restrict__ dst16) {
    __shared__ float cs[16];
    if (threadIdx.x < 16) cs[threadIdx.x] = 0.0f;
    __syncthreads();
    const unsigned tid = blockIdx.x * blockDim.x + threadIdx.x;
    const unsigned stride = gridDim.x * blockDim.x;              // multiple of 16
    const unsigned N = (unsigned)NU * 16u;
    float local = 0.f;
    for (unsigned i = tid; i < N; i += stride) local += src[i];
    atomicAdd(&cs[tid & 15u], local);
    __syncthreads();
    if (threadIdx.x < 16) atomicAdd(dst16 + threadIdx.x, cs[threadIdx.x]);
}

// ---- flat sum of log(x) (precise logf; tiny data) ----
__global__ void k_logsum(const float* __restrict__ src, float* __restrict__ acc, unsigned n) {
    const unsigned tid = blockIdx.x * blockDim.x + threadIdx.x;
    const unsigned stride = gridDim.x * blockDim.x;
    float s = 0.f;
    for (unsigned i = tid; i < n; i += stride) s += logf(src[i]);
#pragma unroll
    for (int off = 16; off >= 1; off >>= 1) s += __shfl_xor(s, off, 32);
    __shared__ float red[8];
    if ((threadIdx.x & 31u) == 0u) red[threadIdx.x >> 5] = s;
    __syncthreads();
    if (threadIdx.x == 0) {
        float b = 0.f;
#pragma unroll
        for (int j = 0; j < 8; ++j) b += red[j];
        atomicAdd(acc, b);
    }
}

// ---- beta-prime log-density sum ----
__global__ void k_beta(const float* __restrict__ p, const float* __restrict__ deg,
                       const float* __restrict__ sigma, float* __restrict__ acc, unsigned n) {
    const unsigned tid = blockIdx.x * blockDim.x + threadIdx.x;
    float term = 0.f;
    if (tid < n) {
        float mu = fminf(fmaxf(deg[tid], 0.001f), 0.999f);
        float sg = sigma[0];
        float a  = mu * (mu * (1.0f - mu) / (sg * sg) - 1.0f);
        float b  = a * (1.0f / mu - 1.0f);
        float pp = p[tid];
        term = (a - 1.0f) * logf(pp) + (b - 1.0f) * log1pf(-pp)
             + lgammaf(a + b) - lgammaf(a) - lgammaf(b) - LOG1000F;
    }
#pragma unroll
    for (int off = 16; off >= 1; off >>= 1) term += __shfl_xor(term, off, 32);
    __shared__ float red[8];
    if ((threadIdx.x & 31u) == 0u) red[threadIdx.x >> 5] = term;
    __syncthreads();
    if (threadIdx.x == 0) {
        float b2 = 0.f;
#pragma unroll
        for (int j = 0; j < 8; ++j) b2 += red[j];
        atomicAdd(acc, b2);
    }
}

// ---- tail: log_softmax, w matrix, final combine (tiny; single thread) ----
__global__ void k_final(const float* __restrict__ logit_pi, const float* __restrict__ logit_psi,
                        const float* __restrict__ ws, float* __restrict__ out) {
    if (threadIdx.x != 0 || blockIdx.x != 0) return;

    float log_pi[16], log_psi[16], row[16], col[16];
    float m = -1e30f;
    for (int k = 0; k < 16; ++k) m = fmaxf(m, logit_pi[k]);
    float se = 0.f;
    for (int k = 0; k < 16; ++k) se += expf(logit_pi[k] - m);
    float lse = m + logf(se);
    for (int k = 0; k < 16; ++k) log_pi[k] = logit_pi[k] - lse;

    m = -1e30f;
    for (int h = 0; h < 16; ++h) m = fmaxf(m, logit_psi[h]);
    se = 0.f;
    for (int h = 0; h < 16; ++h) se += expf(logit_psi[h] - m);
    lse = m + logf(se);
    for (int h = 0; h < 16; ++h) log_psi[h] = logit_psi[h] - lse;

    for (int k = 0; k < 16; ++k) row[k] = log_pi[k]  + ws[1 + k];
    for (int h = 0; h < 16; ++h) col[h] = log_psi[h] + ws[17 + h];

    float M[16][16];
    float S = 0.f;
    for (int k = 0; k < 16; ++k)
        for (int h = 0; h < 16; ++h) { M[k][h] = row[k] + col[h] + 1e-5f; S += M[k][h]; }
    float invS = 1.0f / S;

    float lpu = 0.f, lpi = 0.f;
    for (int k = 0; k < 16; ++k) {
        float wr = 0.f;
        for (int h = 0; h < 16; ++h) wr += M[k][h] * invS;
        lpu += row[k] * wr;
    }
    for (int h = 0; h < 16; ++h) {
        float wc = 0.f;
        for (int k = 0; k < 16; ++k) wc += M[k][h] * invS;
        lpi += col[h] * wc;
    }
    for (int k = 0; k < 16; ++k)
        for (int h = 0; h < 16; ++h) out[6 + k * 16 + h] = M[k][h] * invS;

    const float Nf = (float)NU * (float)NI;
    // ws[0] is 16x the log2-domain sum: undo replication, convert to ln, negate mean
    float loss_pref = -(ws[0] * (LN2F / 16.0f)) / Nf;

    const float a_dir = 0.1f / 16.0f;
    float dir_sum = (a_dir - 1.0f) * ws[33]
                  + (float)NI * (lgammaf(0.1f) - 16.0f * lgammaf(a_dir));
    float neg_dir = -dir_sum;
    float c2 = (neg_dir > 0.0f) ? 1.0f : 0.0f;
    float beta_sum = ws[34];

    float loss = neg_dir * c2 + loss_pref - beta_sum;   // c1 = 0 term vanishes
    out[0] = loss;
    out[1] = -lpu;
    out[2] = -lpi;
    out[3] = neg_dir;
    out[4] = loss_pref;
    out[5] = -beta_sum;
}

extern "C" void kernel_launch(void* const* d_in, const int* in_sizes, int n_in,
                              void* d_out, int out_size, void* d_ws, size_t ws_size,
                              hipStream_t stream) {
    (void)in_sizes; (void)n_in; (void)out_size; (void)ws_size;
    const float* alpha_item = (const float*)d_in[1];
    const float* log_p_user = (const float*)d_in[2];
    const float* log_q_item = (const float*)d_in[3];
    const float* logit_pi   = (const float*)d_in[4];
    const float* logit_psi  = (const float*)d_in[5];
    const float* theta_user = (const float*)d_in[6];
    const float* phi_item   = (const float*)d_in[7];
    const float* sigma_u    = (const float*)d_in[8];
    const float* sigma_i    = (const float*)d_in[9];
    const float* pred       = (const float*)d_in[10];
    const float* udeg       = (const float*)d_in[11];
    const float* ideg       = (const float*)d_in[12];
    const float* ratings    = (const float*)d_in[13];
    float* out = (float*)d_out;
    float* ws  = (float*)d_ws;

    k_init<<<1, 64, 0, stream>>>(ws);
    k_bce<<<BCE_BLOCKS, BCE_THREADS, 0, stream>>>(pred, ratings, ws + 0);
    k_colsum16<<<32, 256, 0, stream>>>(log_p_user, ws + 1);
    k_colsum16<<<32, 256, 0, stream>>>(log_q_item, ws + 17);
    k_logsum<<<64, 256, 0, stream>>>(alpha_item, ws + 33, (unsigned)NU * 16u);
    k_beta<<<32, 256, 0, stream>>>(theta_user, udeg, sigma_u, ws + 34, (unsigned)NU);
    k_beta<<<32, 256, 0, stream>>>(phi_item,  ideg, sigma_i, ws + 34, (unsigned)NI);
    k_final<<<1, 32, 0, stream>>>(logit_pi, logit_psi, ws, out);
}